// GRU_51685636440353
// MI455X (gfx1250) — compile-verified
//
#include <hip/hip_runtime.h>
#include <hip/hip_bf16.h>

// GRU (T=4096, IN=64, H=512) for MI455X / gfx1250.
//
// Strategy:
//  * 32 persistent workgroups; WG g owns hidden channels [16g, 16g+16) and the
//    48 corresponding rows of weight_hh (r/z/n gates), kept resident in LDS
//    (96 KB) so the per-step matvec never touches HBM/L2 for weights.
//  * Per-step cross-WG exchange of the 512-float hidden state via a
//    double-buffered global array in d_ws + a monotonic release/acquire
//    atomic-counter barrier (graph-capture safe, deterministic).
//  * Input-side gate pre-activations are computed 16 timesteps at a time with
//    V_WMMA_F32_16X16X4_F32 (fp32 WMMA): A = X tile (16x4), B = Wih^T tile
//    from LDS (stride-66 padded), C = broadcast bias. 48 dense WMMAs per
//    16-step block per WG, staged into LDS.
//  * Hot matvec uses float2 lane-paired LDS reads (ds_load_b64, all 64 banks
//    touched exactly once per access) and float4 global reads for h.
//
// Workspace layout (d_ws): [0..511] h buf0, [512..1023] h buf1,
//                          [1024] (as uint) barrier counter.  (4100 bytes)

#define T_STEPS 4096
#define IN_DIM  64
#define H_DIM   512
#define NWG     32
#define HC      16              // hidden channels per WG
#define NROWS   48              // 3 gates * HC
#define WIH_STRIDE 66           // padded row stride for Wih slice in LDS

typedef __attribute__((ext_vector_type(2))) float v2f;
typedef __attribute__((ext_vector_type(8))) float v8f;

__device__ __forceinline__ float fast_sigmoid(float x) {
    return 1.0f / (1.0f + __expf(-x));
}
__device__ __forceinline__ float fast_tanh(float x) {
    float e = __expf(-2.0f * x);
    return (1.0f - e) / (1.0f + e);
}

__global__ void gru_init_kernel(const float* __restrict__ h0, float* __restrict__ ws) {
    int i = threadIdx.x;
    if (i < H_DIM) {
        ws[i]         = h0[i];   // buf0 = h_0
        ws[H_DIM + i] = h0[i];   // buf1 (overwritten at step 0 anyway)
    }
    if (i == 0) {
        *((unsigned*)(ws + 2 * H_DIM)) = 0u;   // barrier counter
    }
}

__launch_bounds__(256, 1)
__global__ void gru_scan_kernel(const float* __restrict__ input,      // (T, 64)
                                const float* __restrict__ wih,        // (1536, 64)
                                const float* __restrict__ whh,        // (1536, 512)
                                const float* __restrict__ bias,       // (1536,)
                                const float* __restrict__ bias_n,     // (512,)
                                float* __restrict__ out,              // (T,)
                                float* __restrict__ ws) {
    __shared__ float whh_s[NROWS * H_DIM];         // 96 KB: recurrent weight slice
    __shared__ float wih_s[NROWS * WIH_STRIDE];    // 12.4 KB: input weight slice (padded)
    __shared__ float h_s[H_DIM];                   // current hidden state
    __shared__ float igbuf_s[16 * NROWS];          // input-gate preacts for 16 steps
    __shared__ float pre_s[NROWS];                 // Whh @ h partials
    __shared__ float bias_s[NROWS];
    __shared__ float biasn_s[HC];

    const int g     = blockIdx.x;        // 0..31
    const int tid   = threadIdx.x;       // 0..255
    const int wave  = tid >> 5;          // 0..7
    const int lane  = tid & 31;
    const int hbase = g * HC;

    float* hbuf = ws;                                   // 2 x 512 floats
    unsigned* ctr = (unsigned*)(ws + 2 * H_DIM);

    // ---- Load resident weight slices into LDS (once) -------------------
    // local row lr -> global row: gate(lr/16)*512 + hbase + (lr%16)
    {
        const float4* whh4 = (const float4*)whh;
        float4* whh_s4 = (float4*)whh_s;
        for (int idx = tid; idx < NROWS * (H_DIM / 4); idx += 256) {
            int lr = idx >> 7, c4 = idx & 127;      // 128 float4 per row
            int grow = (lr >> 4) * H_DIM + hbase + (lr & (HC - 1));
            whh_s4[idx] = whh4[(size_t)grow * (H_DIM / 4) + c4];
        }
    }
    for (int idx = tid; idx < NROWS * IN_DIM; idx += 256) {
        int lr = idx >> 6, c = idx & (IN_DIM - 1);
        int grow = (lr >> 4) * H_DIM + hbase + (lr & (HC - 1));
        wih_s[lr * WIH_STRIDE + c] = wih[grow * IN_DIM + c];
    }
    if (tid < NROWS) {
        int grow = (tid >> 4) * H_DIM + hbase + (tid & (HC - 1));
        bias_s[tid] = bias[grow];
    }
    if (tid < HC) biasn_s[tid] = bias_n[hbase + tid];
    __syncthreads();

    // ---- Sequential scan ----------------------------------------------
    for (int t = 0; t < T_STEPS; ++t) {
        // (a) Every 16 steps: compute igates for steps [t, t+16) with fp32 WMMA.
        //     D(16x16) = A(16x4: X rows) x B(4x16: Wih^T cols) + C(bias bcast).
        //     Waves 0..2 each own one 16-column tile (== one gate group).
        if ((t & 15) == 0) {
            if (wave < 3) {
                const int half = lane >> 4;       // 0: K 0..1, 1: K 2..3
                const int l16  = lane & 15;
                float bcol = bias_s[wave * HC + l16];
                v8f acc;
                #pragma unroll
                for (int j = 0; j < 8; ++j) acc[j] = bcol;

                // A: lane (l16,half) holds X[t+l16][4k + 2*half .. +1]
                const float* aptr = input + (size_t)(t + l16) * IN_DIM + 2 * half;
                // B: lane (l16,half) holds Wih^T[4k + 2*half .. +1][col]
                //    = wih_s[col][4k + 2*half .. +1]
                const float* bptr = &wih_s[(wave * HC + l16) * WIH_STRIDE + 2 * half];

                if (t + 16 < T_STEPS)             // prefetch next A block
                    __builtin_prefetch(input + (size_t)(t + 16 + l16) * IN_DIM, 0, 1);

                #pragma unroll
                for (int k = 0; k < 16; ++k) {    // K = 64 in chunks of 4
                    v2f a = *(const v2f*)(aptr + 4 * k);
                    v2f b = *(const v2f*)(bptr + 4 * k);
                    acc = __builtin_amdgcn_wmma_f32_16x16x4_f32(
                        /*neg_a=*/false, a, /*neg_b=*/false, b,
                        /*c_mod=*/(short)0, acc,
                        /*reuse_a=*/false, /*reuse_b=*/false);
                }
                // D layout: VGPR j -> (M=j, N=lane) | (M=j+8, N=lane-16)
                #pragma unroll
                for (int j = 0; j < 8; ++j)
                    igbuf_s[(j + 8 * half) * NROWS + wave * HC + l16] = acc[j];
            }
        }

        // (b) Fetch current hidden state (written by all WGs last step).
        const float4* hsrc4 = (const float4*)(hbuf + (t & 1) * H_DIM);
        if (tid < H_DIM / 4) ((float4*)h_s)[tid] = hsrc4[tid];
        __syncthreads();

        // (c) Recurrent matvec: 48 rows x 512, 8 waves x 6 rows.
        //     Per lane: float2 pair (2*lane + 64*j), covers all 64 LDS banks
        //     exactly once per ds_load_b64 -> conflict-free.
        #pragma unroll
        for (int i = 0; i < 6; ++i) {
            int r = wave * 6 + i;
            const v2f* wr = (const v2f*)&whh_s[r * H_DIM];
            const v2f* hv = (const v2f*)h_s;
            float acc = 0.0f;
            #pragma unroll
            for (int j = 0; j < 8; ++j) {
                v2f w = wr[lane + 32 * j];
                v2f h2 = hv[lane + 32 * j];
                acc = fmaf(w.x, h2.x, fmaf(w.y, h2.y, acc));
            }
            #pragma unroll
            for (int off = 16; off > 0; off >>= 1)
                acc += __shfl_xor(acc, off, 32);
            if (lane == 0) pre_s[r] = acc;
        }
        __syncthreads();

        // (d) Gate math + publish h_new chunk (wave 0, lanes 0..15).
        if (wave == 0) {
            if (lane < HC) {
                const int m = t & 15;
                float ig_r = igbuf_s[m * NROWS + lane];
                float ig_z = igbuf_s[m * NROWS + HC + lane];
                float ig_n = igbuf_s[m * NROWS + 2 * HC + lane];
                float r = fast_sigmoid(ig_r + pre_s[lane]);
                float z = fast_sigmoid(ig_z + pre_s[HC + lane]);
                float n = fast_tanh(ig_n + r * (pre_s[2 * HC + lane] + biasn_s[lane]));
                float hold = h_s[hbase + lane];
                float hnew = n + z * (hold - n);
                hbuf[((t + 1) & 1) * H_DIM + hbase + lane] = hnew;
                if (g == 0 && lane == 0) out[t] = hnew;   // out[t] = h_{t+1}[0]
            }
            __threadfence();                               // release our h chunk
            if (tid == 0)
                __hip_atomic_fetch_add(ctr, 1u, __ATOMIC_RELEASE,
                                       __HIP_MEMORY_SCOPE_AGENT);
        }

        // (e) Device-wide barrier: monotonic counter, acquire on exit.
        if (tid == 0) {
            const unsigned target = (unsigned)NWG * (unsigned)(t + 1);
            while (__hip_atomic_load(ctr, __ATOMIC_ACQUIRE,
                                     __HIP_MEMORY_SCOPE_AGENT) < target)
                __builtin_amdgcn_s_sleep(1);
        }
        __syncthreads();
    }
}

extern "C" void kernel_launch(void* const* d_in, const int* in_sizes, int n_in,
                              void* d_out, int out_size, void* d_ws, size_t ws_size,
                              hipStream_t stream) {
    const float* input  = (const float*)d_in[0];   // (4096, 64)
    const float* h0     = (const float*)d_in[1];   // (512,)
    const float* wih    = (const float*)d_in[2];   // (1536, 64)
    const float* whh    = (const float*)d_in[3];   // (1536, 512)
    const float* bias   = (const float*)d_in[4];   // (1536,)
    const float* bias_n = (const float*)d_in[5];   // (512,)
    float* out = (float*)d_out;                    // (4096,)
    float* ws  = (float*)d_ws;                     // needs >= 4100 bytes

    gru_init_kernel<<<dim3(1), dim3(512), 0, stream>>>(h0, ws);
    gru_scan_kernel<<<dim3(NWG), dim3(256), 0, stream>>>(
        input, wih, whh, bias, bias_n, out, ws);
}